// SoftGumbelQuantizedHologramLayer_21277267984949
// MI455X (gfx1250) — compile-verified
//
#include <hip/hip_runtime.h>
#include <math.h>
#include <stdint.h>

// ---------------- problem constants (from reference) ----------------
#define PI_F        3.14159265358979323846f
#define TWO_PI_F    6.28318530717958647692f
#define INV_2PI_F   0.15915494309189533577f

constexpr int   HImg   = 2048;
constexpr int   WImg   = 2048;
constexpr int   HW     = HImg * WImg;        // 4,194,304 pixels
constexpr int   NWL    = 4;
constexpr int   LLV    = 6;
constexpr int   THREADS = 128;               // 4 waves (wave32)
constexpr int   TILE   = THREADS * 4;        // 512 px per tile
constexpr int   NT     = HW / TILE;          // 8192 tiles
constexpr int   NB     = 2048;               // persistent blocks
constexpr int   PLANES = 16;                 // staged f32 planes per tile
constexpr float SQ3    = 1.73205080756887729f;
constexpr float TAND_F = 0.01f;
constexpr float BASEPT = 2.0e-3f;            // base plane thickness (m)
constexpr float RECT_H = 0.04f;              // rect half width (m)
constexpr float XMIN   = -0.1024f;           // linspace(-DX*W/2, DX*W/2, W)
constexpr float XSTEP  = 0.2048f / 2047.0f;

// ---------------- CDNA5 async global->LDS path ----------------
// The builtin's params are pointers to 16B int vectors in AS(1)/AS(3).
typedef int v4i_t __attribute__((vector_size(4 * sizeof(int))));
typedef __attribute__((address_space(1))) v4i_t* g_v4i_ptr;
typedef __attribute__((address_space(3))) v4i_t* l_v4i_ptr;

#if defined(__has_builtin) && __has_builtin(__builtin_amdgcn_global_load_async_to_lds_b128)
#define ASYNC_LDS_B128(g, l)                                                   \
  __builtin_amdgcn_global_load_async_to_lds_b128(                              \
      (g_v4i_ptr)(uintptr_t)(g),                                               \
      (l_v4i_ptr)(uint32_t)(uintptr_t)(l), 0, 0)
#else
#define ASYNC_LDS_B128(g, l)                                                   \
  asm volatile("global_load_async_to_lds_b128 %0, %1, off"                     \
               :: "v"((uint32_t)(uintptr_t)(l)),                               \
                  "v"((uint64_t)(uintptr_t)(g)) : "memory")
#endif

#if defined(__has_builtin) && __has_builtin(__builtin_amdgcn_s_wait_asynccnt)
#define WAIT_ASYNC(n) do { __builtin_amdgcn_s_wait_asynccnt(n);                \
                           asm volatile("" ::: "memory"); } while (0)
#else
#define WAIT_ASYNC(n) asm volatile("s_wait_asynccnt " #n ::: "memory")
#endif

// Issue 16 async b128 loads: one 16B chunk per plane for this thread's 4 px.
__device__ __forceinline__ void issue_tile(
    int t, int buf, int tid,
    const float* __restrict__ phase, const float* __restrict__ hnoise,
    const float* __restrict__ gum,   const float* __restrict__ freal,
    const float* __restrict__ fimag, float* smem)
{
  const int go = t * TILE + tid * 4;                  // float index into plane
  float* lb = smem + buf * PLANES * TILE + tid * 4;   // this lane's LDS slots
  const float* gp[PLANES];
  gp[0] = phase  + go;
  gp[1] = hnoise + go;
#pragma unroll
  for (int l = 0; l < LLV; ++l) gp[2 + l] = gum + l * HW + go;
#pragma unroll
  for (int w = 0; w < NWL; ++w) {
    gp[8  + w] = freal + w * HW + go;
    gp[12 + w] = fimag + w * HW + go;
  }
#pragma unroll
  for (int pl = 0; pl < PLANES; ++pl)
    ASYNC_LDS_B128(gp[pl], lb + pl * TILE);
}

__global__ __launch_bounds__(THREADS)
void holo_soft_gumbel_kernel(const float* __restrict__ phase,
                             const float* __restrict__ freal,
                             const float* __restrict__ fimag,
                             const float* __restrict__ gum,
                             const float* __restrict__ hnoise,
                             const float* __restrict__ wl,
                             const float* __restrict__ lut,
                             float* __restrict__ out)
{
  __shared__ float smem[2 * PLANES * TILE];           // 64 KB, double buffered
  const int tid = threadIdx.x;
  const int bid = blockIdx.x;

  // Small uniform tables -> registers, plus per-wavelength exponent coeffs.
  float wlv[NWL], lutv[LLV], Ae[NWL], Bp[NWL];
#pragma unroll
  for (int w = 0; w < NWL; ++w) wlv[w] = wl[w];
#pragma unroll
  for (int l = 0; l < LLV; ++l) lutv[l] = lut[l];
  const float dwl = fminf(fminf(wlv[0], wlv[1]), fminf(wlv[2], wlv[3]));
#pragma unroll
  for (int w = 0; w < NWL; ++w) {
    const float k = TWO_PI_F / wlv[w];
    Ae[w] = -0.5f * k * TAND_F * SQ3;   // loss  = exp(Ae*hb)
    Bp[w] = -k * (SQ3 - 1.0f);          // theta = Bp*hb
  }
  const float hscale = dwl / (TWO_PI_F * (SQ3 - 1.0f));

  int t   = bid;
  int buf = 0;
  issue_tile(t, buf, tid, phase, hnoise, gum, freal, fimag, smem);

  while (true) {
    const int tn = t + NB;
    if (tn < NT) {
      issue_tile(tn, buf ^ 1, tid, phase, hnoise, gum, freal, fimag, smem);
      WAIT_ASYNC(16);   // 32 in flight; oldest 16 (current tile) complete
    } else {
      WAIT_ASYNC(0);
    }

    const float* sb = smem + buf * PLANES * TILE + tid * 4;
    const float4 ph4 = *(const float4*)(sb + 0 * TILE);
    const float4 hn4 = *(const float4*)(sb + 1 * TILE);
    float4 g4[LLV];
#pragma unroll
    for (int l = 0; l < LLV; ++l) g4[l] = *(const float4*)(sb + (2 + l) * TILE);
    float4 fr4[NWL], fi4[NWL];
#pragma unroll
    for (int w = 0; w < NWL; ++w) {
      fr4[w] = *(const float4*)(sb + (8  + w) * TILE);
      fi4[w] = *(const float4*)(sb + (12 + w) * TILE);
    }

    const int   p0 = t * TILE + tid * 4;
    const int   yr = p0 >> 11;                 // row (W == 2048)
    const float fy = fmaf((float)yr, XSTEP, XMIN);
    const float my = (fabsf(fy) <= RECT_H) ? 1.0f : 0.0f;

    float ore[NWL][4], oim[NWL][4];
#pragma unroll
    for (int j = 0; j < 4; ++j) {
      const int   xc   = (p0 + j) & (WImg - 1);
      const float fx   = fmaf((float)xc, XSTEP, XMIN);
      const float mask = ((fabsf(fx) <= RECT_H) ? 1.0f : 0.0f) * my;

      // wrapped = (phase + pi) mod 2pi - pi
      float a = ((const float*)&ph4)[j] + PI_F;
      a -= TWO_PI_F * floorf(a * INV_2PI_F);
      const float wrapped = a - PI_F;

      // scores + gumbel logits, track max for stable softmax
      float logitv[LLV];
      float mx = -3.402823e38f;
#pragma unroll
      for (int l = 0; l < LLV; ++l) {
        float d = wrapped - lutv[l] + PI_F;
        d -= TWO_PI_F * floorf(d * INV_2PI_F);
        d -= PI_F;
        const float z  = 1.0f / (1.0f + __expf(-5.0f * d));
        const float sc = 4.0f * z * (1.0f - z);
        const float lg = (sc + ((const float*)&g4[l])[j]) * (1.0f / 3.0f);
        logitv[l] = lg;
        mx = fmaxf(mx, lg);
      }
      float ssum = 0.0f, q = 0.0f;
#pragma unroll
      for (int l = 0; l < LLV; ++l) {
        const float e = __expf(logitv[l] - mx);
        ssum += e;
        q = fmaf(e, lutv[l], q);
      }
      q /= ssum;

      // height -> hb (base thickness + fab noise)
      float qm = q - TWO_PI_F * floorf(q * INV_2PI_F);
      const float hb = fmaf(qm, hscale, ((const float*)&hn4)[j] + BASEPT);

#pragma unroll
      for (int w = 0; w < NWL; ++w) {
        const float loss = __expf(Ae[w] * hb);
        float s, c;
        __sincosf(Bp[w] * hb, &s, &c);
        const float pr  = loss * c;
        const float pim = loss * s;
        const float fr  = ((const float*)&fr4[w])[j];
        const float fim = ((const float*)&fi4[w])[j];
        ore[w][j] = mask * (fr * pr - fim * pim);
        oim[w][j] = mask * (fr * pim + fim * pr);
      }
    }

    // interleaved complex64 output: two float4 stores per wavelength
#pragma unroll
    for (int w = 0; w < NWL; ++w) {
      float4* dst = (float4*)(out + 2ll * ((long long)w * HW + p0));
      dst[0] = make_float4(ore[w][0], oim[w][0], ore[w][1], oim[w][1]);
      dst[1] = make_float4(ore[w][2], oim[w][2], ore[w][3], oim[w][3]);
    }

    if (tn >= NT) break;
    t = tn;
    buf ^= 1;
  }
}

extern "C" void kernel_launch(void* const* d_in, const int* in_sizes, int n_in,
                              void* d_out, int out_size, void* d_ws, size_t ws_size,
                              hipStream_t stream) {
  const float* phase  = (const float*)d_in[0];  // [1,1,H,W]
  const float* freal  = (const float*)d_in[1];  // [1,4,H,W]
  const float* fimag  = (const float*)d_in[2];  // [1,4,H,W]
  const float* gum    = (const float*)d_in[3];  // [1,6,H,W]
  const float* hnoise = (const float*)d_in[4];  // [H,W]
  const float* wl     = (const float*)d_in[5];  // [4]
  const float* lut    = (const float*)d_in[6];  // [6]
  float* out = (float*)d_out;                   // complex64 [1,4,H,W] interleaved

  (void)in_sizes; (void)n_in; (void)out_size; (void)d_ws; (void)ws_size;

  hipLaunchKernelGGL(holo_soft_gumbel_kernel, dim3(NB), dim3(THREADS), 0, stream,
                     phase, freal, fimag, gum, hnoise, wl, lut, out);
}